// ResidualFSQ_31061203485237
// MI455X (gfx1250) — compile-verified
//
#include <hip/hip_runtime.h>
#include <hip/hip_bf16.h>

// ---------------------------------------------------------------------------
// Residual FSQ (8 stages, levels [8,5,5,5,5,5]) for MI455X / gfx1250.
//
// Kernel 1 (stats): one workgroup per (b,d) row; whole 256KB row lives in LDS
//   (CDNA5 gives 320KB/WGP). Row is DMA'd in with the Tensor Data Mover
//   (tensor_load_to_lds + s_wait_tensorcnt), then 8 reduce+update stages run
//   entirely out of LDS. Only per-stage (mean,std) scalars hit HBM.
// Kernel 2 (emit): elementwise replay given the stats; streams x once and
//   writes quantized_out and indices. ~180MB total HBM traffic -> ~8us floor.
// ---------------------------------------------------------------------------

typedef unsigned int v4u __attribute__((ext_vector_type(4)));
typedef int          v8i __attribute__((ext_vector_type(8)));
typedef int          v4i __attribute__((ext_vector_type(4)));

#define LROW      65536      // time length per (b,d) row
#define NB        32
#define ND        6
#define NQ        8
#define THREADS_A 1024
#define CHUNK_A   16         // float4 elements per thread in kernel A (16*1024*4 = 65536)

// One FSQ stage on a single scalar; updates residual in place, returns bracket.
// Used identically by both kernels so the replay matches bitwise.
__device__ __forceinline__ float stage_step(float& res, float mean, float stdv,
                                            float g, float be, float lm1) {
  float norm = g * (res - mean) / stdv + be;               // InvertibleLayerNorm fwd
  float zc   = fminf(fmaxf(norm, -1.f), 1.f);
  float br   = floorf(lm1 * (zc + 1.f) * 0.5f + 0.5f);     // FSQ bracket
  br         = fminf(fmaxf(br, 0.f), lm1);
  float code = (2.f / lm1) * br - 1.f;                     // quantized (STE fwd value)
  float qt   = (code - be) / g * stdv + mean;              // inverse norm
  res        = res - qt;
  return br;
}

// ---------------------------------------------------------------------------
// Kernel A: per-row statistics via LDS-resident residual.
// grid = 192 (= 32*6), block = 1024, dynamic LDS = (65536 + 8) floats.
// stats layout: stats[(row*8 + q)*2 + {0:mean, 1:std}]
// ---------------------------------------------------------------------------
extern "C" __global__ void __launch_bounds__(THREADS_A, 1)
fsq_stats_kernel(const float* __restrict__ x,
                 const float* __restrict__ gamma,
                 const float* __restrict__ beta,
                 float* __restrict__ stats) {
  extern __shared__ float smem[];           // [0,65536): row; [65536]: sum; [65537]: sumsq
  float* ssum = smem + LROW;
  float* ssq  = smem + LROW + 1;
  const int row = blockIdx.x;               // row = b*6 + d
  const int d   = row % ND;
  const int tid = threadIdx.x;
  float4* row4  = reinterpret_cast<float4*>(smem);

  // ---- Stage the 256KB row into LDS with the Tensor Data Mover (wave 0) ----
#if __has_builtin(__builtin_amdgcn_tensor_load_to_lds) && __has_builtin(__builtin_amdgcn_s_wait_tensorcnt)
  if (tid < 32) {
    unsigned long long ga = (unsigned long long)(const void*)(x + (size_t)row * LROW);
    unsigned ldsAddr = (unsigned)(size_t)(void*)smem;
    // D# group 0: count=1, lds_addr, global_addr[56:0], type=2 ("image")
    v4u g0;
    g0[0] = 1u;
    g0[1] = ldsAddr;
    g0[2] = (unsigned)(ga & 0xFFFFFFFFull);
    g0[3] = ((unsigned)((ga >> 32) & 0x1FFFFFFull)) | (2u << 30);
    // D# group 1: data_size=8B; 1-D tile of 32768 x 8B = 256KB (tile_dim0 fits 16 bits)
    v8i g1 = (v8i)0;
    g1[0] = (int)(3u << 16);          // workgroup_mask=0, data_size=3 (8 bytes)
    g1[1] = (int)(32768u << 16);      // tensor_dim0 = 32768 (bits 63:48 = low16)
    g1[2] = (int)(1u << 16);          // tensor_dim1 = 1     (bits 95:80)
    g1[3] = (int)(32768u << 16);      // tile_dim0   = 32768 (bits 127:112)
    g1[4] = 1;                        // tile_dim1   = 1, tile_dim2 = 0
    g1[5] = 32768;                    // tensor_dim0_stride (bits 191:160)
    g1[6] = (int)(32768u << 16);      // tensor_dim1_stride low16 (bits 223:208)
    g1[7] = 0;
    v4i g2 = (v4i)0;
    v4i g3 = (v4i)0;
#if defined(__clang_major__) && (__clang_major__ >= 23)
    v8i g4 = (v8i)0;
    __builtin_amdgcn_tensor_load_to_lds(g0, g1, g2, g3, g4, 0);
#else
    __builtin_amdgcn_tensor_load_to_lds(g0, g1, g2, g3, 0);
#endif
    __builtin_amdgcn_s_wait_tensorcnt(0);
  }
#else
  // Fallback: cooperative vectorized copy global -> LDS.
  for (int j = 0; j < CHUNK_A; ++j) {
    int i = j * THREADS_A + tid;
    row4[i] = reinterpret_cast<const float4*>(x + (size_t)row * LROW)[i];
  }
#endif
  __syncthreads();

  // ---- Soft clamp: residual_0 = clamp * tanh(x / clamp), clamp = 1 + 1/(L-1) ----
  const float lm1d = (d == 0) ? 7.f : 4.f;
  const float cl   = 1.f + 1.f / lm1d;
  for (int j = 0; j < CHUNK_A; ++j) {
    int i = j * THREADS_A + tid;     // interleaved: conflict-free across lanes
    float4 v = row4[i];
    v.x = cl * tanhf(v.x / cl);
    v.y = cl * tanhf(v.y / cl);
    v.z = cl * tanhf(v.z / cl);
    v.w = cl * tanhf(v.w / cl);
    row4[i] = v;
  }
  // Each thread owns its elements exclusively; barriers below only guard scratch.

  for (int q = 0; q < NQ; ++q) {
    // partial sum / sumsq over this thread's 64 elements
    float s = 0.f, s2 = 0.f;
#pragma unroll
    for (int j = 0; j < CHUNK_A; ++j) {
      float4 v = row4[j * THREADS_A + tid];
      s  += v.x + v.y + v.z + v.w;
      s2 += v.x * v.x + v.y * v.y + v.z * v.z + v.w * v.w;
    }
    // wave32 reduction
#pragma unroll
    for (int off = 16; off > 0; off >>= 1) {
      s  += __shfl_down(s,  off, 32);
      s2 += __shfl_down(s2, off, 32);
    }
    __syncthreads();                               // prev-stage scratch reads done
    if (tid == 0) { *ssum = 0.f; *ssq = 0.f; }
    __syncthreads();
    if ((tid & 31) == 0) { atomicAdd(ssum, s); atomicAdd(ssq, s2); }
    __syncthreads();

    float mean = *ssum * (1.f / (float)LROW);
    float var  = *ssq  * (1.f / (float)LROW) - mean * mean;   // biased var
    float stdv = sqrtf(var + 1e-5f);
    if (tid == 0) {
      stats[(row * NQ + q) * 2 + 0] = mean;
      stats[(row * NQ + q) * 2 + 1] = stdv;
    }
    float g  = gamma[q * ND + d];
    float be = beta[q * ND + d];
    // residual update in LDS
    for (int j = 0; j < CHUNK_A; ++j) {
      int i = j * THREADS_A + tid;
      float4 v = row4[i];
      stage_step(v.x, mean, stdv, g, be, lm1d);
      stage_step(v.y, mean, stdv, g, be, lm1d);
      stage_step(v.z, mean, stdv, g, be, lm1d);
      stage_step(v.w, mean, stdv, g, be, lm1d);
      row4[i] = v;
    }
  }
}

// ---------------------------------------------------------------------------
// Kernel B: elementwise replay; writes quantized_out and indices.
// grid = (65536/(4*256)=64, 32), block = 256; each thread does 4 consecutive t.
// ---------------------------------------------------------------------------
extern "C" __global__ void __launch_bounds__(256, 1)
fsq_emit_kernel(const float* __restrict__ x,
                const float* __restrict__ gamma,
                const float* __restrict__ beta,
                const float* __restrict__ stats,
                float* __restrict__ outq,
                float* __restrict__ outidx) {
  const int b  = blockIdx.y;
  const int t0 = (blockIdx.x * 256 + threadIdx.x) * 4;

  const float lm1c[ND] = {7.f, 4.f, 4.f, 4.f, 4.f, 4.f};
  const float basc[ND] = {1.f, 8.f, 40.f, 200.f, 1000.f, 5000.f};

  float rr[ND][4];   // residual
  float xo[ND][4];   // soft-clamped input (for quantized_out = xo - rr_final)
#pragma unroll
  for (int d = 0; d < ND; ++d) {
    const float cl = 1.f + 1.f / lm1c[d];
    float4 v = *reinterpret_cast<const float4*>(x + (((size_t)(b * ND + d)) << 16) + t0);
    float tmp[4] = {v.x, v.y, v.z, v.w};
#pragma unroll
    for (int k = 0; k < 4; ++k) {
      float c = cl * tanhf(tmp[k] / cl);
      rr[d][k] = c;
      xo[d][k] = c;
    }
  }

  float ia[NQ][4];   // index per stage per element (exact integers in fp32)
#pragma unroll
  for (int q = 0; q < NQ; ++q) {
#pragma unroll
    for (int k = 0; k < 4; ++k) ia[q][k] = 0.f;
#pragma unroll
    for (int d = 0; d < ND; ++d) {
      float mean = stats[((b * ND + d) * NQ + q) * 2 + 0];
      float stdv = stats[((b * ND + d) * NQ + q) * 2 + 1];
      float g    = gamma[q * ND + d];
      float be   = beta[q * ND + d];
#pragma unroll
      for (int k = 0; k < 4; ++k) {
        float br = stage_step(rr[d][k], mean, stdv, g, be, lm1c[d]);
        ia[q][k] += br * basc[d];
      }
    }
  }

  // indices: layout [b, t, q] -> for each t, 8 consecutive q values
  float* ip = outidx + ((size_t)((b << 16) + t0)) * NQ;
#pragma unroll
  for (int k = 0; k < 4; ++k) {
    float4 a, c;
    a.x = ia[0][k]; a.y = ia[1][k]; a.z = ia[2][k]; a.w = ia[3][k];
    c.x = ia[4][k]; c.y = ia[5][k]; c.z = ia[6][k]; c.w = ia[7][k];
    *reinterpret_cast<float4*>(ip + (size_t)k * NQ)     = a;
    *reinterpret_cast<float4*>(ip + (size_t)k * NQ + 4) = c;
  }

  // quantized_out = x_clamped - residual_final
#pragma unroll
  for (int d = 0; d < ND; ++d) {
    float4 o;
    o.x = xo[d][0] - rr[d][0];
    o.y = xo[d][1] - rr[d][1];
    o.z = xo[d][2] - rr[d][2];
    o.w = xo[d][3] - rr[d][3];
    *reinterpret_cast<float4*>(outq + (((size_t)(b * ND + d)) << 16) + t0) = o;
  }
}

// ---------------------------------------------------------------------------
extern "C" void kernel_launch(void* const* d_in, const int* in_sizes, int n_in,
                              void* d_out, int out_size, void* d_ws, size_t ws_size,
                              hipStream_t stream) {
  const float* x     = (const float*)d_in[0];   // [32, 6, 65536]
  const float* gamma = (const float*)d_in[1];   // [8, 6]
  const float* beta  = (const float*)d_in[2];   // [8, 6]

  float* stats  = (float*)d_ws;                 // [192][8][2] = 12 KB
  float* outq   = (float*)d_out;                // quantized_out, 12.58M floats
  float* outidx = outq + (size_t)NB * ND * LROW;// indices as exact fp32 ints

  const size_t smemA = (size_t)(LROW + 8) * sizeof(float);  // 256KB row + scratch
  (void)hipFuncSetAttribute(reinterpret_cast<const void*>(&fsq_stats_kernel),
                            hipFuncAttributeMaxDynamicSharedMemorySize, (int)smemA);

  fsq_stats_kernel<<<NB * ND, THREADS_A, smemA, stream>>>(x, gamma, beta, stats);
  fsq_emit_kernel<<<dim3(LROW / (4 * 256), NB), 256, 0, stream>>>(
      x, gamma, beta, stats, outq, outidx);
}